// RNN_classification_50783693308001
// MI455X (gfx1250) — compile-verified
//
#include <hip/hip_runtime.h>
#include <hip/hip_bf16.h>
#include <math.h>

#define S_LEN 2048
#define BATCH 256
#define HID   128

typedef float v2f __attribute__((ext_vector_type(2)));
typedef float v4f __attribute__((ext_vector_type(4)));
typedef float v8f __attribute__((ext_vector_type(8)));

// ---------------------------------------------------------------------------
// Phase A: xw[b,s,:] = x[b,s,:] @ W_ih^T + b_ih
// M = 16 batch rows (fixed s), N = 128 (8 n-tiles), K = 128.
// x tile (16 rows x 512B) staged into LDS with CDNA5 async-to-LDS b128
// (each instruction = one fully contiguous 512B row), then WMMA A-layout
// chunks are read from LDS (row stride 132 floats -> conflict-free banks).
// Output written pre-permuted into WMMA C-layout for phase B:
//   scratch[((s*16 + btile)*8 + ntile)*256 + lane*8 + r]
// xw is a >L2-sized single-use stream -> non-temporal stores.
// ---------------------------------------------------------------------------
__global__ __launch_bounds__(128) void rnn_input_proj(
    const float* __restrict__ x, const float* __restrict__ Wih,
    const float* __restrict__ bih, float* __restrict__ xw)
{
    __shared__ float xtile[4][16][132];        // one 16x128 tile per wave (+pad)

    const int lane  = threadIdx.x & 31;
    const int wave  = threadIdx.x >> 5;        // 4 waves per block
    const int s     = blockIdx.x * 4 + wave;   // [0, 2048)
    const int bt    = blockIdx.y;              // [0, 16) batch tile
    const int n     = lane & 15;               // A-row (M) / B-col (N) / C-col
    const int khalf = lane >> 4;               // 0: K={0,1}  1: K={2,3} in chunk

    // --- async stage: 16 rows, each row = 32 lanes x 16B contiguous ---------
#pragma unroll
    for (int r = 0; r < 16; ++r) {
        const float* src = x + ((size_t)(bt * 16 + r) * S_LEN + s) * HID + lane * 4;
        unsigned     dst = (unsigned)(size_t)&xtile[wave][r][lane * 4];
        asm volatile("global_load_async_to_lds_b128 %0, %1, off th:TH_LOAD_NT"
                     :: "v"(dst), "v"(src) : "memory");
    }
    asm volatile("s_wait_asynccnt 0" ::: "memory");   // LDS data visible

    // --- A operand (16x4 f32 per chunk) from LDS ----------------------------
    // lane holds rows M = lane%16, components K = 4c + 2*khalf + {0,1}
    v2f a[32];
#pragma unroll
    for (int c = 0; c < 32; ++c)
        a[c] = *(const v2f*)(&xtile[wave][n][4 * c + 2 * khalf]);

    float* outbase = xw + (((size_t)s * 16 + bt) * 8) * 256;

#pragma unroll
    for (int nt = 0; nt < 8; ++nt) {
        // B operand (4x16 f32): B[k][col] = W_ih[col][k], col = nt*16 + n
        // W_ih is 64KB -> L2/WGP$ resident, direct global loads are fine.
        const float* wrow = Wih + (size_t)(nt * 16 + n) * HID;
        const float  bias = bih[nt * 16 + n];   // C col = lane%16
        v8f acc = {};
#pragma unroll
        for (int c = 0; c < 32; ++c) {
            v2f b = *(const v2f*)(wrow + 4 * c + 2 * khalf);
            acc = __builtin_amdgcn_wmma_f32_16x16x4_f32(
                false, a[c], false, b, (short)0, acc, false, false);
        }
        float* o = outbase + nt * 256 + lane * 8;
        v4f lo = {acc[0] + bias, acc[1] + bias, acc[2] + bias, acc[3] + bias};
        v4f hi = {acc[4] + bias, acc[5] + bias, acc[6] + bias, acc[7] + bias};
        __builtin_nontemporal_store(lo, (v4f*)(o));
        __builtin_nontemporal_store(hi, (v4f*)(o + 4));
    }
}

// ---------------------------------------------------------------------------
// Phase B: sequential recurrence. One workgroup per 16 batch rows (16 WGs),
// 8 waves = 8 N-tiles of h_new. W_hh^T slice resident in VGPRs for all steps;
// h shared via LDS (row stride 132 floats -> conflict-free across 64 banks).
// K-accumulation split over 4 accumulators to cut the WMMA RAW chain to 8.
// xw stream read non-temporally + prefetched 4 steps ahead.
// ---------------------------------------------------------------------------
__global__ __launch_bounds__(256) void rnn_recurrence(
    const float* __restrict__ xw, const float* __restrict__ Whh,
    const float* __restrict__ bhh, float* __restrict__ hout)
{
    __shared__ float hbuf[16][132];            // 16 batch rows x 128 cols (+pad)

    const int lane  = threadIdx.x & 31;
    const int nt    = threadIdx.x >> 5;        // wave = N-tile
    const int bt    = blockIdx.x;              // batch tile
    const int n     = lane & 15;
    const int khalf = lane >> 4;

    // Resident B operand: B[k][col] = W_hh[col][k], col = nt*16 + n  (64 VGPRs)
    const float* wrow = Whh + (size_t)(nt * 16 + n) * HID;
    v2f bw[32];
#pragma unroll
    for (int c = 0; c < 32; ++c)
        bw[c] = *(const v2f*)(wrow + 4 * c + 2 * khalf);
    const float bias = bhh[nt * 16 + n];

    // h0 = 0
    for (int i = threadIdx.x; i < 16 * 132; i += 256)
        (&hbuf[0][0])[i] = 0.0f;
    __syncthreads();

    const float* xwp = xw + ((size_t)bt * 8 + nt) * 256 + lane * 8;
    const size_t sstride = (size_t)16 * 8 * 256;   // floats per timestep

    v8f hv = {};
    for (int s = 0; s < S_LEN; ++s) {
        const float* p = xwp + (size_t)s * sstride;
        v4f xlo = __builtin_nontemporal_load((const v4f*)(p));
        v4f xhi = __builtin_nontemporal_load((const v4f*)(p + 4));
        if (s + 4 < S_LEN)
            __builtin_prefetch(p + 4 * sstride, 0, 0);   // global_prefetch_b8

        // A operand chunks: h rows M = lane%16 from LDS
        v2f ha[32];
#pragma unroll
        for (int c = 0; c < 32; ++c)
            ha[c] = *(const v2f*)(&hbuf[n][4 * c + 2 * khalf]);
        __syncthreads();   // all waves finished reading h_{s-1}

        v8f acc0 = {xlo[0] + bias, xlo[1] + bias, xlo[2] + bias, xlo[3] + bias,
                    xhi[0] + bias, xhi[1] + bias, xhi[2] + bias, xhi[3] + bias};
        v8f acc1 = {}, acc2 = {}, acc3 = {};
#pragma unroll
        for (int c = 0; c < 32; c += 4) {
            acc0 = __builtin_amdgcn_wmma_f32_16x16x4_f32(
                false, ha[c + 0], false, bw[c + 0], (short)0, acc0, false, false);
            acc1 = __builtin_amdgcn_wmma_f32_16x16x4_f32(
                false, ha[c + 1], false, bw[c + 1], (short)0, acc1, false, false);
            acc2 = __builtin_amdgcn_wmma_f32_16x16x4_f32(
                false, ha[c + 2], false, bw[c + 2], (short)0, acc2, false, false);
            acc3 = __builtin_amdgcn_wmma_f32_16x16x4_f32(
                false, ha[c + 3], false, bw[c + 3], (short)0, acc3, false, false);
        }
#pragma unroll
        for (int r = 0; r < 8; ++r)
            hv[r] = tanhf(acc0[r] + acc1[r] + acc2[r] + acc3[r]);

        // h_s back to LDS in C-layout: row = r + 8*khalf, col = nt*16 + n
#pragma unroll
        for (int r = 0; r < 8; ++r)
            hbuf[r + 8 * khalf][nt * 16 + n] = hv[r];
        __syncthreads();   // all waves finished writing h_s
    }

    // final h -> d_out (256 x 128, fp32)
#pragma unroll
    for (int r = 0; r < 8; ++r)
        hout[(size_t)(bt * 16 + r + 8 * khalf) * HID + nt * 16 + n] = hv[r];
}

extern "C" void kernel_launch(void* const* d_in, const int* in_sizes, int n_in,
                              void* d_out, int out_size, void* d_ws, size_t ws_size,
                              hipStream_t stream) {
    const float* x   = (const float*)d_in[0];
    const float* Wih = (const float*)d_in[1];
    const float* bih = (const float*)d_in[2];
    const float* Whh = (const float*)d_in[3];
    const float* bhh = (const float*)d_in[4];
    float*       xw  = (float*)d_ws;          // needs 2048*256*128*4 = 256 MiB

    dim3 g1(S_LEN / 4, BATCH / 16);           // 512 x 16 blocks, 4 waves each
    rnn_input_proj<<<g1, 128, 0, stream>>>(x, Wih, bih, xw);
    rnn_recurrence<<<BATCH / 16, 256, 0, stream>>>(xw, Whh, bhh, (float*)d_out);
}